// PCBLoss_10514079941540
// MI455X (gfx1250) — compile-verified
//
#include <hip/hip_runtime.h>
#include <math.h>

// Problem constants (from reference)
#define A_     3
#define C_     6
#define CH_    11          // 5 + C
#define H_     160
#define W_     160
#define HW_    (H_*W_)     // 25600
#define N_     (A_*HW_)    // 76800
#define B_     32
#define M_     50
#define LAMBDA_COORD 5.0f
#define LAMBDA_NOOBJ 0.5f
#define LN2_F  0.69314718055994530942f

// K2 decomposition: per (b,a) plane split into CHUNKS chunks; WAVES waves per block
#define K2_THREADS 256
#define K2_CHUNKS  8
#define K2_WAVES   (K2_THREADS/32)
#define SLOTS_PER_B (A_*K2_CHUNKS*K2_WAVES)   // 192 partial slots per batch

// ws float layout:
//   [32..63]   S_obj_neg_logp   (per batch)
//   [64..95]   S_obj_neg_log1mp (per batch)
//   [96..127]  S_coord_sse      (per batch)
//   [128..159] S_class_ce       (per batch)
//   [160..191] n_resp (as int)
//   [256..256+32*192) per-wave partials of S_all_neg_log1mp
#define WS_NRESP_OFF 160
#define WS_PART_OFF  256

typedef __attribute__((ext_vector_type(2))) float v2f;
typedef __attribute__((ext_vector_type(8))) float v8f;

// ds_swizzle group-of-32 XOR patterns: offset = (xor_mask<<10) | (or<<5) | and_mask
#define SWZ_XOR8  0x201F
#define SWZ_XOR4  0x101F
#define SWZ_XOR2  0x081F
#define SWZ_XOR1  0x041F
#define SWZ_ADD(v, pat)                                                     \
  (v) += __int_as_float(__builtin_amdgcn_ds_swizzle(__float_as_int(v), (pat)))

// Exact wave32 sum of (x,y) across all 32 lanes using V_WMMA_F32_16X16X4_F32.
// A = ones(16x4)  =>  D[i][j] = colsum_j(B).  The 64 B register slots map
// bijectively onto the 4x16 matrix, so Sum_j colsum(j) == Sum of all 64 slots,
// independent of the exact VGPR<->matrix layout. The 16 distinct column sums
// live (replicated) in d[0] of lanes 0-15 and 16-31; 4 XOR swizzles finish.
__device__ __forceinline__ float wave_sum32(float x, float y) {
  v2f a;  a[0] = 1.0f; a[1] = 1.0f;
  v2f bv; bv[0] = x;   bv[1] = y;
  v8f c = {};
  c = __builtin_amdgcn_wmma_f32_16x16x4_f32(
      /*neg_a=*/false, a, /*neg_b=*/false, bv,
      /*c_mod=*/(short)0, c, /*reuse_a=*/false, /*reuse_b=*/false);
  float v = c[0];
  SWZ_ADD(v, SWZ_XOR8);
  SWZ_ADD(v, SWZ_XOR4);
  SWZ_ADD(v, SWZ_XOR2);
  SWZ_ADD(v, SWZ_XOR1);
  return v;   // total in every lane
}

// Fast natural log via the hardware transcendental v_log_f32 (log2).
// Valid here: arguments are in (0.02, 0.98), far from denormals/zero.
__device__ __forceinline__ float fast_ln(float x) {
  return __builtin_amdgcn_logf(x) * LN2_F;
}

// ---------------------------------------------------------------------------
// Kernel 1: per-batch target assignment (scatter emulation) + object-cell math
// One block per batch, 64 threads (>= M_=50). Deterministic LDS reduction.
// ---------------------------------------------------------------------------
__global__ void assign_obj_kernel(const float* __restrict__ pred,
                                  const float* __restrict__ tgt,
                                  float* __restrict__ ws_f,
                                  int*   __restrict__ ws_n) {
  __shared__ int   s_idx[M_];
  __shared__ int   s_cls[M_];
  __shared__ float s_box[M_][4];
  __shared__ int   s_win[M_];
  __shared__ float s_lp[M_], s_l1[M_], s_sse[M_], s_ce[M_];

  const int b = blockIdx.x;
  const int j = threadIdx.x;

  if (j < M_) {
    const float* t = tgt + ((size_t)b * M_ + j) * 6;
    float t0 = t[0], t1 = t[1], t2 = t[2], t3 = t[3], t4 = t[4], t5 = t[5];
    bool valid = (t5 > 0.0f);
    int gx = min((int)(t1 * (float)W_), W_ - 1);
    int gy = min((int)(t2 * (float)H_), H_ - 1);
    s_idx[j] = valid ? (gy * W_ + gx) : -1;
    s_cls[j] = (int)t0;
    s_box[j][0] = t1; s_box[j][1] = t2; s_box[j][2] = t3; s_box[j][3] = t4;
    s_win[j] = 0;
    s_lp[j] = 0.0f; s_l1[j] = 0.0f; s_sse[j] = 0.0f; s_ce[j] = 0.0f;
  }
  __syncthreads();

  if (j < M_ && s_idx[j] >= 0) {
    // last-writer-wins for the box; class = min class over the duplicate group
    // (matches argmax over the union one-hot).
    bool winner = true;
    int  mc = s_cls[j];
    for (int k = 0; k < M_; ++k) {
      if (k != j && s_idx[k] == s_idx[j]) {
        if (k > j) winner = false;
        mc = min(mc, s_cls[k]);
      }
    }
    if (winner) {
      s_win[j] = 1;
      const int cell = s_idx[j];
      const float* base = pred + (size_t)b * A_ * CH_ * HW_ + cell;  // anchor 0
      float sse = 0.0f;
      #pragma unroll
      for (int c = 0; c < 4; ++c) {
        float d = base[(size_t)c * HW_] - s_box[j][c];
        sse += d * d;
      }
      float conf = base[(size_t)4 * HW_];
      float lp = fmaxf(logf(conf),    -100.0f);
      float l1 = fmaxf(log1pf(-conf), -100.0f);
      float lg[C_];
      float m = -1e30f;
      #pragma unroll
      for (int c = 0; c < C_; ++c) {
        lg[c] = base[(size_t)(5 + c) * HW_];
        m = fmaxf(m, lg[c]);
      }
      float se = 0.0f;
      #pragma unroll
      for (int c = 0; c < C_; ++c) se += expf(lg[c] - m);
      float ce = (m + logf(se)) - lg[mc];
      s_lp[j] = -lp;  s_l1[j] = -l1;  s_sse[j] = sse;  s_ce[j] = ce;
    }
  }
  __syncthreads();

  if (j == 0) {   // fixed-order reduction -> deterministic
    float Sop = 0.0f, So1 = 0.0f, Sco = 0.0f, Sce = 0.0f;
    int n = 0;
    for (int k = 0; k < M_; ++k) {
      Sop += s_lp[k]; So1 += s_l1[k]; Sco += s_sse[k]; Sce += s_ce[k];
      n += s_win[k];
    }
    ws_f[32 + b]  = Sop;
    ws_f[64 + b]  = So1;
    ws_f[96 + b]  = Sco;
    ws_f[128 + b] = Sce;
    ws_n[b] = n;
  }
}

// ---------------------------------------------------------------------------
// Kernel 2: stream ONLY the confidence channel (ch 4) of every (b,a) plane.
// grid = (B*A, K2_CHUNKS), 256 threads. float4 (b128) coalesced loads.
// log via v_log_f32 (~4 VALU/element) to stay under the 23.3 TB/s roof.
// Per-wave totals via WMMA reduction; each wave writes a unique partial slot
// (no atomics -> bitwise deterministic).
// ---------------------------------------------------------------------------
__global__ void conf_plane_kernel(const float* __restrict__ pred,
                                  float* __restrict__ ws_f) {
  const int plane = blockIdx.x;          // b*A + a
  const int b = plane / A_;
  const int a = plane - b * A_;
  const float4* __restrict__ src =
      (const float4*)(pred + ((size_t)plane * CH_ + 4) * HW_);
  const int q4 = HW_ / 4;                          // 6400 float4 per plane
  const int chunk4 = q4 / K2_CHUNKS;               // 800
  const int i0 = blockIdx.y * chunk4;

  float acc0 = 0.0f, acc1 = 0.0f;
  for (int i = i0 + threadIdx.x; i < i0 + chunk4; i += K2_THREADS) {
    float4 v = src[i];
    acc0 += fmaxf(fast_ln(1.0f - v.x), -100.0f)
          + fmaxf(fast_ln(1.0f - v.z), -100.0f);
    acc1 += fmaxf(fast_ln(1.0f - v.y), -100.0f)
          + fmaxf(fast_ln(1.0f - v.w), -100.0f);
  }
  // all 256 threads converged here -> EXEC all ones for WMMA
  float tot = wave_sum32(acc0, acc1);   // wave sum of clamped log1p(-conf)
  const int wave = threadIdx.x >> 5;
  if ((threadIdx.x & 31) == 0) {
    int slot = (a * K2_CHUNKS + blockIdx.y) * K2_WAVES + wave;   // [0,192)
    ws_f[WS_PART_OFF + b * SLOTS_PER_B + slot] = -tot;           // sum of -log1p
  }
}

// ---------------------------------------------------------------------------
// Kernel 3: finalize. One wave of 32 lanes, lane b = batch b.
// ---------------------------------------------------------------------------
__global__ void finalize_kernel(const float* __restrict__ ws_f,
                                const int*   __restrict__ ws_n,
                                float* __restrict__ out) {
  const int b = threadIdx.x;   // 0..31, B_ == 32

  float S_all = 0.0f;          // sum over ALL N cells of -log1p(-conf) (clamped)
  const float* p = ws_f + WS_PART_OFF + b * SLOTS_PER_B;
  for (int s = 0; s < SLOTS_PER_B; ++s) S_all += p[s];

  float Sop = ws_f[32 + b];    // sum_obj -log(conf)
  float So1 = ws_f[64 + b];    // sum_obj -log1p(-conf)
  float Sco = ws_f[96 + b];    // sum_obj coord SSE
  float Sce = ws_f[128 + b];   // sum_obj CE
  int   nr  = ws_n[b];
  bool  has = (nr > 0);        // == has_valid (any valid target sets a cell)
  float fn  = (float)nr;

  float obj_conf   = Sop / fmaxf(fn, 1.0f);
  float noobj_conf = (S_all - So1) / fmaxf((float)(N_ - nr), 1.0f);
  float conf_b  = has ? (obj_conf + LAMBDA_NOOBJ * noobj_conf)
                      : (S_all / (float)N_);          // empty_conf == mean
  float coord_b = has ? (Sco / fmaxf(4.0f * fn, 1.0f)) : 0.0f;
  float class_b = has ? (Sce / fmaxf(fn, 1.0f))        : 0.0f;

  // three exact cross-batch sums via WMMA wave reduction
  float coordL = wave_sum32(coord_b, 0.0f);
  float confL  = wave_sum32(conf_b,  0.0f);
  float classL = wave_sum32(class_b, 0.0f);

  if (b == 0) {
    out[0] = LAMBDA_COORD * coordL + confL + classL;
    out[1] = coordL;
    out[2] = confL;
    out[3] = classL;
  }
}

extern "C" void kernel_launch(void* const* d_in, const int* in_sizes, int n_in,
                              void* d_out, int out_size, void* d_ws, size_t ws_size,
                              hipStream_t stream) {
  const float* pred = (const float*)d_in[0];   // (32,3,11,160,160) f32
  const float* tgt  = (const float*)d_in[1];   // (32,50,6) f32
  float* ws_f = (float*)d_ws;
  int*   ws_n = (int*)d_ws + WS_NRESP_OFF;
  float* out  = (float*)d_out;                 // 4 scalars

  assign_obj_kernel<<<dim3(B_), dim3(64), 0, stream>>>(pred, tgt, ws_f, ws_n);
  conf_plane_kernel<<<dim3(B_ * A_, K2_CHUNKS), dim3(K2_THREADS), 0, stream>>>(pred, ws_f);
  finalize_kernel<<<dim3(1), dim3(32), 0, stream>>>(ws_f, ws_n, out);
}